// O3TensorProduct_84232898609799
// MI455X (gfx1250) — compile-verified
//
#include <hip/hip_runtime.h>
#include <hip/hip_bf16.h>

// O3 tensor product (e3nn-style), MUL=128.
// Six N x 128 @ 128 x 128 fp32 GEMMs on the fp32 WMMA path
// (v_wmma_f32_16x16x4_f32), weights pre-swizzled into LDS fragment layout,
// x1 stored k-major in LDS so every A/B fragment is one contiguous
// ds_load_b64 at a DS-immediate offset (no repacking movs in the loop).

#define TP_MUL 128
constexpr int BLOCK  = 256;   // 8 waves (wave32)
constexpr int WAVES  = 8;
constexpr int MROWS  = 16;    // WMMA M
constexpr int KCH    = 32;    // 128 / 4 K-chunks

typedef float v2f __attribute__((ext_vector_type(2)));
typedef float v8f __attribute__((ext_vector_type(8)));

// LDS layout in floats:
//  [0, 65536)           : weight fragments  Wf[wtile][mat][kk][lane][j]
//                         (per-wave 4-matrix window = 32KB -> DS imm offsets)
//  [65536, 65536+8192)  : x block [16][512]; per row: [0..128)=x0,
//                         [128+k*128+u]=x1[u][k]  (k-major transpose)
//  [73728, 73728+64)    : y block [16][4]
constexpr int WEIGHT_FLOATS = 8 * 4 * KCH * 64;       // 65536
constexpr int XOFF          = WEIGHT_FLOATS;          // 65536
constexpr int YOFF          = XOFF + MROWS * 512;     // 73728
constexpr size_t LDS_BYTES  = (size_t)(YOFF + 64) * sizeof(float); // 295168 B <= 320KB WGP LDS

__global__ void __launch_bounds__(BLOCK, 1)
O3TensorProduct_84232898609799_kernel(const float* __restrict__ x,
                                      const float* __restrict__ y,
                                      const float* __restrict__ w000,
                                      const float* __restrict__ w110,
                                      const float* __restrict__ w011,
                                      const float* __restrict__ w101,
                                      const float* __restrict__ bias,
                                      float* __restrict__ out,
                                      int nblocks)
{
    extern __shared__ float smem[];
    const int tid       = threadIdx.x;
    const int lane      = tid & 31;
    const int wave      = tid >> 5;      // 0..7 == output-column tile
    const int colInTile = lane & 15;
    const int half      = lane >> 4;     // 0 or 1 (K pair select for A frag)

    // ---- Stage all four weight matrices into B-fragment LDS layout ----
    // B frag layout for 16x16x4 f32: VGPR0 lanes0-15 = K-row 0, lanes16-31 = K-row 2;
    // VGPR1 = K-rows 1,3.  Store so each lane's v2f is one contiguous b64.
    for (int idx = tid; idx < 4 * 128 * 128; idx += BLOCK) {
        const int mat = idx >> 14;          // 0:W000 1:W110 2:W011 3:W101
        const int rem = idx & 16383;
        const int u   = rem >> 7;           // K index
        const int w   = rem & 127;          // output column
        const float* wp = (mat == 0) ? w000 : (mat == 1) ? w110
                         : (mat == 2) ? w011 : w101;
        const float val = wp[u * 128 + w];
        const int wt = w >> 4, col = w & 15;
        const int kk = u >> 2, r = u & 3;
        const int l  = ((r >> 1) << 4) + col;
        const int j  = r & 1;
        smem[(((wt * 4 + mat) * KCH + kk) << 6) + l * 2 + j] = val;
    }
    const float bval = bias[wave * 16 + colInTile];
    __syncthreads();

    const float inv_s3 = 0.57735026918962576451f;
    // Single per-lane base for this wave's 32KB weight window; all fragment
    // loads below are base + compile-time-constant offsets (DS imm field).
    const float* wbase = smem + ((wave * 4 * KCH) << 6) + lane * 2;

    for (int rb = blockIdx.x; rb < nblocks; rb += gridDim.x) {
        const int n0 = rb * MROWS;
        __syncthreads();   // previous iteration's LDS reads complete

        // ---- Stage x block: x0 as-is (512 float4), x1 transposed k-major ----
        {
            #pragma unroll
            for (int i = 0; i < 2; ++i) {
                const int j = tid + i * BLOCK;        // 0..511 float4s of x0
                const int row = j >> 5, c4 = j & 31;
                ((float4*)(smem + XOFF + row * 512))[c4] =
                    ((const float4*)(x + (size_t)(n0 + row) * 512))[c4];
            }
            #pragma unroll
            for (int i = 0; i < 8; ++i) {
                const int j = tid + i * BLOCK;        // 0..2047 (row,u) triples
                const int row = j >> 7, u = j & 127;
                const float* src = x + (size_t)(n0 + row) * 512 + 128 + 3 * u;
                const float e0 = src[0], e1 = src[1], e2 = src[2];
                float* dst = smem + XOFF + row * 512 + 128 + u;
                dst[0]   = e0;     // x1[u][0] -> k=0 plane
                dst[128] = e1;     // k=1 plane
                dst[256] = e2;     // k=2 plane
            }
            if (tid < MROWS)
                ((float4*)(smem + YOFF))[tid] = ((const float4*)(y + (size_t)n0 * 4))[tid];
        }
        __syncthreads();

        // Prefetch next row block's x into cache while we compute this one
        // (lowers to global_prefetch_b8; one cacheline per lane address).
        {
            const int rbn = rb + gridDim.x;
            if (rbn < nblocks) {
                const float4* xnext = (const float4*)(x + (size_t)rbn * MROWS * 512);
                #pragma unroll
                for (int i = 0; i < 8; ++i)
                    __builtin_prefetch((const void*)(xnext + tid + i * BLOCK), 0, 1);
            }
        }

        // A-fragment source row for this lane is m = colInTile.
        // Per-lane base: all four A-side fragments are contiguous v2f at
        // imm offsets {o, 128+o, 256+o, 384+o} floats.
        const float* xb   = smem + XOFF + colInTile * 512 + half * 2;
        const float* yrow = smem + YOFF + colInTile * 4;
        const float y1r0 = yrow[1], y1r1 = yrow[2], y1r2 = yrow[3];

        v8f accA = {}, accB = {}, accC = {};
        v8f accD0 = {}, accD1 = {}, accD2 = {};

        #pragma unroll 8
        for (int kk = 0; kk < KCH; ++kk) {
            const int o = kk * 4;
            v2f a0 = *(const v2f*)(xb + o);          // x0 pair
            v2f d0 = *(const v2f*)(xb + 128 + o);    // x1[.][0] pair
            v2f d1 = *(const v2f*)(xb + 256 + o);    // x1[.][1] pair
            v2f d2 = *(const v2f*)(xb + 384 + o);    // x1[.][2] pair
            v2f a1 = d0 * y1r0 + d1 * y1r1 + d2 * y1r2;   // packed fma

            // Weight fragments: base + imm offsets within the 32KB window.
            v2f bA = *(const v2f*)(wbase + ((0 * KCH + kk) << 6));  // W000
            v2f bB = *(const v2f*)(wbase + ((1 * KCH + kk) << 6));  // W110
            v2f bC = *(const v2f*)(wbase + ((2 * KCH + kk) << 6));  // W011
            v2f bD = *(const v2f*)(wbase + ((3 * KCH + kk) << 6));  // W101

            accA  = __builtin_amdgcn_wmma_f32_16x16x4_f32(false, a0, false, bA, (short)0, accA,  false, false);
            accB  = __builtin_amdgcn_wmma_f32_16x16x4_f32(false, a1, false, bB, (short)0, accB,  false, false);
            accC  = __builtin_amdgcn_wmma_f32_16x16x4_f32(false, a0, false, bC, (short)0, accC,  false, false);
            accD0 = __builtin_amdgcn_wmma_f32_16x16x4_f32(false, d0, false, bD, (short)0, accD0, false, false);
            accD1 = __builtin_amdgcn_wmma_f32_16x16x4_f32(false, d1, false, bD, (short)0, accD1, false, false);
            accD2 = __builtin_amdgcn_wmma_f32_16x16x4_f32(false, d2, false, bD, (short)0, accD2, false, false);
        }

        // ---- Epilogue: C/D layout is lane L, VGPR r -> row (L/16)*8+r, col wave*16+L%16
        const int cgl = wave * 16 + colInTile;
        #pragma unroll
        for (int r = 0; r < 8; ++r) {
            const int m = half * 8 + r;
            const float* yr = smem + YOFF + m * 4;
            const float yy0 = yr[0], yy1 = yr[1], yy2 = yr[2], yy3 = yr[3];
            const float o0  = yy0 * accA[r] + inv_s3 * accB[r] + bval;
            const float c1  = inv_s3 * accC[r];
            const float g0  = inv_s3 * yy0;
            float* orow = out + (size_t)(n0 + m) * 512;
            orow[cgl] = o0;
            float* o1p = orow + 128 + 3 * cgl;
            o1p[0] = c1 * yy1 + g0 * accD0[r];
            o1p[1] = c1 * yy2 + g0 * accD1[r];
            o1p[2] = c1 * yy3 + g0 * accD2[r];
        }
    }
}

extern "C" void kernel_launch(void* const* d_in, const int* in_sizes, int n_in,
                              void* d_out, int out_size, void* d_ws, size_t ws_size,
                              hipStream_t stream) {
    const float* x    = (const float*)d_in[0];
    const float* y    = (const float*)d_in[1];
    const float* w000 = (const float*)d_in[2];
    const float* w011 = (const float*)d_in[3];
    const float* w101 = (const float*)d_in[4];
    const float* w110 = (const float*)d_in[5];
    const float* bias = (const float*)d_in[6];
    float* out = (float*)d_out;

    const int n       = in_sizes[0] / 512;   // N = 100000
    const int nblocks = n / MROWS;           // 6250 (N is a multiple of 16)

    // Allow the large dynamic LDS allocation (288.25 KB of the 320 KB WGP LDS).
    (void)hipFuncSetAttribute((const void*)O3TensorProduct_84232898609799_kernel,
                              hipFuncAttributeMaxDynamicSharedMemorySize,
                              (int)LDS_BYTES);

    int grid = nblocks < 2048 ? nblocks : 2048;
    if (grid < 1) grid = 1;
    O3TensorProduct_84232898609799_kernel<<<grid, BLOCK, LDS_BYTES, stream>>>(
        x, y, w000, w110, w011, w101, bias, out, nblocks);
}